// MyModel_83846351552867
// MI455X (gfx1250) — compile-verified
//
#include <hip/hip_runtime.h>
#include <math.h>

// ---------------------------------------------------------------------------
// LN-BiLSTM-CRF (NER tagger) for MI455X / gfx1250.
//   - All matmuls via V_WMMA_F32_16X16X4_F32 (fp32-faithful to reference).
//   - Recurrent state + gate buffers live entirely in WGP LDS (320KB/WGP).
//   - Activation staging uses gfx1250 async LDS engine:
//       global_load_async_to_lds_b128 / global_store_async_from_lds_b128
//       completed with s_wait_asynccnt (ASYNCcnt).
//   - Word BiLSTM: 1 workgroup per direction, both layers fused per timestep.
//   - Char BiLSTM: 512 workgroups (256 seq tiles x 2 dirs).
// ---------------------------------------------------------------------------

typedef __attribute__((ext_vector_type(2))) float v2f;
typedef __attribute__((ext_vector_type(8))) float v8f;

#define NSEQ   4096      // BS*SL char sequences
#define BS     32
#define SLEN   128
#define CLEN   16
#define NCLS   18

__device__ __forceinline__ v8f wmma4(v2f a, v2f b, v8f c) {
  // D = A(16x4,f32) * B(4x16,f32) + C(16x16,f32)
  return __builtin_amdgcn_wmma_f32_16x16x4_f32(false, a, false, b, (short)0, c,
                                               false, false);
}
__device__ __forceinline__ float sigm(float x) { return 1.0f / (1.0f + expf(-x)); }

// ---- gfx1250 async LDS copies (ASYNCcnt path) -----------------------------
// LDS byte address = low 32 bits of the generic pointer to __shared__ memory
// (ISA 10.2, LDS aperture).
__device__ __forceinline__ void async_g2l_b128(void* lds_dst, const void* gsrc) {
  unsigned int l = (unsigned int)(unsigned long long)lds_dst;
  asm volatile("global_load_async_to_lds_b128 %0, %1, off"
               :: "v"(l), "v"(gsrc) : "memory");
}
__device__ __forceinline__ void async_l2g_b128(void* gdst, const void* lds_src) {
  unsigned int l = (unsigned int)(unsigned long long)lds_src;
  asm volatile("global_store_async_from_lds_b128 %0, %1, off"
               :: "v"(gdst), "v"(l) : "memory");
}
__device__ __forceinline__ void async_wait0() {
  asm volatile("s_wait_asynccnt 0x0" ::: "memory");
}

struct CellP {
  const float *Wx, *bx, *Wh, *bh;
  const float *lnxg, *lnxb, *lnhg, *lnhb, *lncg, *lncb;
};

// ===========================================================================
// Char BiLSTM: din=64, h=64, 4h=256, T=16. Grid (256, 2): 16 seqs/block, y=dir.
// Writes final hidden state to hcat[dir*NSEQ + n][64].
// ===========================================================================
__global__ void __launch_bounds__(256)
char_lstm_kernel(const float* __restrict__ c_emb, const int* __restrict__ cidx,
                 CellP cf, CellP cb, float* __restrict__ hcat) {
  __shared__ __align__(16) float xs[16][68];   // row stride 272B (16B mult.)
  __shared__ float hsb[16][68];
  __shared__ float cs[16][64];
  __shared__ float gxm[16][260];
  __shared__ float ghm[16][260];
  __shared__ float red[4][16][16];
  __shared__ float stat[16][4];
  __shared__ float cred[2][16][16];
  __shared__ float cstat[16][2];

  const int tid = threadIdx.x;
  const int dir = blockIdx.y;           // 0=fwd, 1=bwd
  const CellP P = dir ? cb : cf;
  const int n0 = blockIdx.x * 16;

  for (int i = tid; i < 16 * 68; i += 256) hsb[i / 68][i % 68] = 0.f;
  for (int i = tid; i < 16 * 64; i += 256) cs[i / 64][i % 64] = 0.f;
  __syncthreads();

  const int lane = tid & 31;
  const int wave = tid >> 5;
  const int ml = lane & 15;             // A row / B col / D col
  const int kb = (lane >> 4) * 2;       // k base within 4-chunk
  const int rr = tid >> 4;              // LN phase: row 0..15
  const int pp = tid & 15;              // LN phase: partition 0..15

  for (int t = 0; t < CLEN; ++t) {
    const int tc = dir ? (CLEN - 1 - t) : t;
    // stage x_t = c_emb[c[n][tc]] : 16 rows x 16 float4 via async LDS DMA
    {
      int r = tid >> 4, q = tid & 15;   // 256 threads = 16x16 float4 tiles
      int cid = cidx[(n0 + r) * CLEN + tc];
      async_g2l_b128(&xs[r][q * 4], &c_emb[cid * 64 + q * 4]);
      async_wait0();
    }
    __syncthreads();

    // gx = x @ Wx^T + bx ; gh = h @ Wh^T + bh   (both [16,256], K=64)
    for (int jt = wave; jt < 16; jt += 8) {
      const int ncol = jt * 16 + ml;
      v8f ax = {}; v8f ah = {};
      const float* wxr = P.Wx + ncol * 64;
      const float* whr = P.Wh + ncol * 64;
      for (int k4 = 0; k4 < 16; ++k4) {
        const int k0 = k4 * 4 + kb;
        v2f a, b;
        a[0] = xs[ml][k0]; a[1] = xs[ml][k0 + 1];
        b[0] = wxr[k0];    b[1] = wxr[k0 + 1];
        ax = wmma4(a, b, ax);
        v2f a2, b2;
        a2[0] = hsb[ml][k0]; a2[1] = hsb[ml][k0 + 1];
        b2[0] = whr[k0];     b2[1] = whr[k0 + 1];
        ah = wmma4(a2, b2, ah);
      }
      const float bxv = P.bx[ncol], bhv = P.bh[ncol];
#pragma unroll
      for (int v = 0; v < 8; ++v) {
        int row = v + (lane >> 4) * 8;
        gxm[row][ncol] = ax[v] + bxv;
        ghm[row][ncol] = ah[v] + bhv;
      }
    }
    __syncthreads();

    // LayerNorm stats over the 256-wide gate rows (gx and gh separately)
    {
      float sx = 0, qx = 0, sh = 0, qh = 0;
      for (int cth = 0; cth < 16; ++cth) {
        int col = pp * 16 + cth;
        float g = gxm[rr][col]; sx += g; qx += g * g;
        float g2 = ghm[rr][col]; sh += g2; qh += g2 * g2;
      }
      red[0][rr][pp] = sx; red[1][rr][pp] = qx;
      red[2][rr][pp] = sh; red[3][rr][pp] = qh;
    }
    __syncthreads();
    if (pp == 0) {
      float s0 = 0, s1 = 0, s2 = 0, s3 = 0;
      for (int i = 0; i < 16; ++i) {
        s0 += red[0][rr][i]; s1 += red[1][rr][i];
        s2 += red[2][rr][i]; s3 += red[3][rr][i];
      }
      float mx = s0 / 256.f, mh = s2 / 256.f;
      stat[rr][0] = mx; stat[rr][1] = rsqrtf(s1 / 256.f - mx * mx + 1e-5f);
      stat[rr][2] = mh; stat[rr][3] = rsqrtf(s3 / 256.f - mh * mh + 1e-5f);
    }
    __syncthreads();

    // gates + c update (each thread: 4 cells of its row)
    float cnv[4], ogv[4];
    {
      const float mx = stat[rr][0], rx = stat[rr][1];
      const float mh = stat[rr][2], rh = stat[rr][3];
      float s = 0, q = 0;
      for (int cc = 0; cc < 4; ++cc) {
        int j = pp * 4 + cc;
        float gi = P.lnxb[j]       + P.lnxg[j]       * (gxm[rr][j]       - mx) * rx
                 + P.lnhb[j]       + P.lnhg[j]       * (ghm[rr][j]       - mh) * rh;
        float gf = P.lnxb[64 + j]  + P.lnxg[64 + j]  * (gxm[rr][64 + j]  - mx) * rx
                 + P.lnhb[64 + j]  + P.lnhg[64 + j]  * (ghm[rr][64 + j]  - mh) * rh;
        float go = P.lnxb[128 + j] + P.lnxg[128 + j] * (gxm[rr][128 + j] - mx) * rx
                 + P.lnhb[128 + j] + P.lnhg[128 + j] * (ghm[rr][128 + j] - mh) * rh;
        float gu = P.lnxb[192 + j] + P.lnxg[192 + j] * (gxm[rr][192 + j] - mx) * rx
                 + P.lnhb[192 + j] + P.lnhg[192 + j] * (ghm[rr][192 + j] - mh) * rh;
        float cn = cs[rr][j] * sigm(gf) + tanhf(gu) * sigm(gi);
        cnv[cc] = cn; ogv[cc] = go; s += cn; q += cn * cn;
      }
      cred[0][rr][pp] = s; cred[1][rr][pp] = q;
    }
    __syncthreads();
    if (pp == 0) {
      float s = 0, q = 0;
      for (int i = 0; i < 16; ++i) { s += cred[0][rr][i]; q += cred[1][rr][i]; }
      float m = s / 64.f;
      cstat[rr][0] = m; cstat[rr][1] = rsqrtf(q / 64.f - m * m + 1e-5f);
    }
    __syncthreads();
    {
      const float mc = cstat[rr][0], rc = cstat[rr][1];
      for (int cc = 0; cc < 4; ++cc) {
        int j = pp * 4 + cc;
        float hn = sigm(ogv[cc]) * tanhf(P.lncb[j] + P.lncg[j] * (cnv[cc] - mc) * rc);
        hsb[rr][j] = hn;
        cs[rr][j] = cnv[cc];
      }
    }
    __syncthreads();
  }

  for (int i = tid; i < 16 * 64; i += 256) {
    int r = i >> 6, col = i & 63;
    hcat[(dir * NSEQ + n0 + r) * 64 + col] = hsb[r][col];
  }
}

// ===========================================================================
// Build word-LSTM input X[t][b][0:384] = [w_emb[s[b][t]] | e_c[b][t]]
// e_c replicates the reference's concat-axis-1 reshape quirk via hcat rows.
// ===========================================================================
__global__ void __launch_bounds__(256)
build_x_kernel(const float* __restrict__ w_emb, const int* __restrict__ sidx,
               const float* __restrict__ hcat, float* __restrict__ X) {
  int idx = blockIdx.x * 256 + threadIdx.x;
  if (idx >= SLEN * BS * 384) return;
  int f = idx % 384;
  int b = (idx / 384) % BS;
  int t = idx / (384 * BS);
  float v;
  if (f < 256) {
    v = w_emb[(size_t)sidx[b * SLEN + t] * 256 + f];
  } else {
    int fc = f - 256;
    int n = b * SLEN + t;
    v = hcat[(2 * n + (fc >> 6)) * 64 + (fc & 63)];
  }
  X[idx] = v;
}

// ===========================================================================
// Word LSTM half-step for rows [mt*16, mt*16+16): gates, LN, state update.
// 4h = 768 (48 col-tiles over 24 waves), cell width 192.
// ===========================================================================
__device__ __forceinline__ void word_layer_halfstep(
    int tid, int mt, const float* xs, int xstride, int Kx, float* hs,
    int hstride, float* csA, const CellP P,
    float (*gxm)[772], float (*ghm)[772], float (*red)[16][48],
    float (*stat)[4], float (*cred)[16][48], float (*cstat)[2]) {
  const int lane = tid & 31;
  const int wave = tid >> 5;
  const int ml = lane & 15;
  const int kb = (lane >> 4) * 2;
  const int rowA = mt * 16 + ml;

  for (int jt = wave; jt < 48; jt += 24) {
    const int ncol = jt * 16 + ml;
    v8f ax = {}; v8f ah = {};
    const float* wxr = P.Wx + (size_t)ncol * Kx;
    for (int k4 = 0; k4 < Kx / 4; ++k4) {
      int k0 = k4 * 4 + kb;
      v2f a, b;
      a[0] = xs[rowA * xstride + k0]; a[1] = xs[rowA * xstride + k0 + 1];
      b[0] = wxr[k0];                 b[1] = wxr[k0 + 1];
      ax = wmma4(a, b, ax);
    }
    const float* whr = P.Wh + (size_t)ncol * 192;
    for (int k4 = 0; k4 < 48; ++k4) {
      int k0 = k4 * 4 + kb;
      v2f a, b;
      a[0] = hs[rowA * hstride + k0]; a[1] = hs[rowA * hstride + k0 + 1];
      b[0] = whr[k0];                 b[1] = whr[k0 + 1];
      ah = wmma4(a, b, ah);
    }
    const float bxs = P.bx[ncol], bhs = P.bh[ncol];
#pragma unroll
    for (int v = 0; v < 8; ++v) {
      int row = v + (lane >> 4) * 8;
      gxm[row][ncol] = ax[v] + bxs;
      ghm[row][ncol] = ah[v] + bhs;
    }
  }
  __syncthreads();

  const int rr = tid / 48;   // 0..15
  const int pp = tid % 48;   // 0..47
  {
    float sx = 0, qx = 0, sh = 0, qh = 0;
    for (int cth = 0; cth < 16; ++cth) {
      int col = pp * 16 + cth;
      float g = gxm[rr][col]; sx += g; qx += g * g;
      float g2 = ghm[rr][col]; sh += g2; qh += g2 * g2;
    }
    red[0][rr][pp] = sx; red[1][rr][pp] = qx;
    red[2][rr][pp] = sh; red[3][rr][pp] = qh;
  }
  __syncthreads();
  if (pp == 0) {
    float s0 = 0, s1 = 0, s2 = 0, s3 = 0;
    for (int i = 0; i < 48; ++i) {
      s0 += red[0][rr][i]; s1 += red[1][rr][i];
      s2 += red[2][rr][i]; s3 += red[3][rr][i];
    }
    float mx = s0 / 768.f, mh = s2 / 768.f;
    stat[rr][0] = mx; stat[rr][1] = rsqrtf(s1 / 768.f - mx * mx + 1e-5f);
    stat[rr][2] = mh; stat[rr][3] = rsqrtf(s3 / 768.f - mh * mh + 1e-5f);
  }
  __syncthreads();

  float cnv[4], ogv[4];
  const int grow = mt * 16 + rr;
  {
    const float mx = stat[rr][0], rx = stat[rr][1];
    const float mh = stat[rr][2], rh = stat[rr][3];
    float s = 0, q = 0;
    for (int cc = 0; cc < 4; ++cc) {
      int j = pp * 4 + cc;
      float gi = P.lnxb[j]       + P.lnxg[j]       * (gxm[rr][j]       - mx) * rx
               + P.lnhb[j]       + P.lnhg[j]       * (ghm[rr][j]       - mh) * rh;
      float gf = P.lnxb[192 + j] + P.lnxg[192 + j] * (gxm[rr][192 + j] - mx) * rx
               + P.lnhb[192 + j] + P.lnhg[192 + j] * (ghm[rr][192 + j] - mh) * rh;
      float go = P.lnxb[384 + j] + P.lnxg[384 + j] * (gxm[rr][384 + j] - mx) * rx
               + P.lnhb[384 + j] + P.lnhg[384 + j] * (ghm[rr][384 + j] - mh) * rh;
      float gu = P.lnxb[576 + j] + P.lnxg[576 + j] * (gxm[rr][576 + j] - mx) * rx
               + P.lnhb[576 + j] + P.lnhg[576 + j] * (ghm[rr][576 + j] - mh) * rh;
      float cn = csA[grow * 192 + j] * sigm(gf) + tanhf(gu) * sigm(gi);
      cnv[cc] = cn; ogv[cc] = go; s += cn; q += cn * cn;
    }
    cred[0][rr][pp] = s; cred[1][rr][pp] = q;
  }
  __syncthreads();
  if (pp == 0) {
    float s = 0, q = 0;
    for (int i = 0; i < 48; ++i) { s += cred[0][rr][i]; q += cred[1][rr][i]; }
    float m = s / 192.f;
    cstat[rr][0] = m; cstat[rr][1] = rsqrtf(q / 192.f - m * m + 1e-5f);
  }
  __syncthreads();
  {
    const float mc = cstat[rr][0], rc = cstat[rr][1];
    for (int cc = 0; cc < 4; ++cc) {
      int j = pp * 4 + cc;
      float hn = sigm(ogv[cc]) * tanhf(P.lncb[j] + P.lncg[j] * (cnv[cc] - mc) * rc);
      hs[grow * hstride + j] = hn;
      csA[grow * 192 + j] = cnv[cc];
    }
  }
  __syncthreads();
}

// ===========================================================================
// Word 2-layer LN-LSTM, one direction per workgroup. Grid(2), block 768.
// All h/c/gate state LDS-resident (~267KB of the 320KB WGP LDS).
// x_t staged with async LDS DMA; h1 written out with async LDS->global.
// ===========================================================================
__global__ void __launch_bounds__(768)
word_lstm_kernel(const float* __restrict__ Xg, CellP f0, CellP f1, CellP b0,
                 CellP b1, float* __restrict__ outF, float* __restrict__ outB) {
  __shared__ __align__(16) float xs[32][388];  // row stride 1552B (16B mult.)
  __shared__ __align__(16) float h0[32][196];
  __shared__ float c0[32][192];
  __shared__ __align__(16) float h1[32][196];  // row stride 784B (16B mult.)
  __shared__ float c1[32][192];
  __shared__ float gxm[16][772];
  __shared__ float ghm[16][772];
  __shared__ float red[4][16][48];
  __shared__ float stat[16][4];
  __shared__ float cred[2][16][48];
  __shared__ float cstat[16][2];

  const int tid = threadIdx.x;
  const int dir = blockIdx.x;
  const CellP L0 = dir ? b0 : f0;
  const CellP L1 = dir ? b1 : f1;
  float* outp = dir ? outB : outF;

  for (int i = tid; i < 32 * 196; i += 768) {
    h0[i / 196][i % 196] = 0.f; h1[i / 196][i % 196] = 0.f;
  }
  for (int i = tid; i < 32 * 192; i += 768) {
    c0[i / 192][i % 192] = 0.f; c1[i / 192][i % 192] = 0.f;
  }
  __syncthreads();

  for (int t = 0; t < SLEN; ++t) {
    const int tc = dir ? (SLEN - 1 - t) : t;
    const int tn = dir ? (tc - 1) : (tc + 1);
    // stage x_t [32][384] via async LDS DMA (3072 float4's over 768 threads)
    for (int i = tid; i < 32 * 96; i += 768) {
      int r = i / 96, q = i % 96;
      async_g2l_b128(&xs[r][q * 4], &Xg[(tc * 32 + r) * 384 + q * 4]);
      if (tn >= 0 && tn < SLEN)  // prefetch next timestep toward L2
        __builtin_prefetch(&Xg[(tn * 32 + r) * 384 + q * 4], 0, 1);
    }
    async_wait0();
    __syncthreads();
    // layer 0 (Kx = 384)
    word_layer_halfstep(tid, 0, &xs[0][0], 388, 384, &h0[0][0], 196, &c0[0][0],
                        L0, gxm, ghm, red, stat, cred, cstat);
    word_layer_halfstep(tid, 1, &xs[0][0], 388, 384, &h0[0][0], 196, &c0[0][0],
                        L0, gxm, ghm, red, stat, cred, cstat);
    // layer 1 (input = h0, Kx = 192)
    word_layer_halfstep(tid, 0, &h0[0][0], 196, 192, &h1[0][0], 196, &c1[0][0],
                        L1, gxm, ghm, red, stat, cred, cstat);
    word_layer_halfstep(tid, 1, &h0[0][0], 196, 192, &h1[0][0], 196, &c1[0][0],
                        L1, gxm, ghm, red, stat, cred, cstat);
    // write out h1 [32][192] via async LDS->global (1536 float4's)
    for (int i = tid; i < 32 * 48; i += 768) {
      int r = i / 48, q = i % 48;
      async_l2g_b128(&outp[(tc * 32 + r) * 192 + q * 4], &h1[r][q * 4]);
    }
    async_wait0();   // h1 must be fully drained before next-step update
    __syncthreads();
  }
}

// ===========================================================================
// Final MLP: e = relu(feat @ W1^T + b1) @ W2^T + b2   -> E[4096][18]
// ===========================================================================
__global__ void __launch_bounds__(256)
mlp_kernel(const float* __restrict__ OUTF, const float* __restrict__ OUTB,
           const float* __restrict__ W1, const float* __restrict__ b1,
           const float* __restrict__ W2, const float* __restrict__ b2,
           float* __restrict__ E) {
  __shared__ __align__(16) float feat[16][388];
  __shared__ float hid[16][260];
  const int tid = threadIdx.x;
  const int n0 = blockIdx.x * 16;
  // stage features via async LDS DMA (1536 float4's over 256 threads)
  for (int i = tid; i < 16 * 96; i += 256) {
    int r = i / 96, f4 = (i % 96) * 4;
    int n = n0 + r, b = n >> 7, t = n & 127;
    const float* src = (f4 < 192) ? &OUTF[(t * 32 + b) * 192 + f4]
                                  : &OUTB[(t * 32 + b) * 192 + (f4 - 192)];
    async_g2l_b128(&feat[r][f4], src);
  }
  async_wait0();
  __syncthreads();
  const int lane = tid & 31, wave = tid >> 5;
  const int ml = lane & 15, kb = (lane >> 4) * 2;
  for (int jt = wave; jt < 16; jt += 8) {
    int ncol = jt * 16 + ml;
    v8f acc = {};
    const float* wr = W1 + (size_t)ncol * 384;
    for (int k4 = 0; k4 < 96; ++k4) {
      int k0 = k4 * 4 + kb;
      v2f a, b;
      a[0] = feat[ml][k0]; a[1] = feat[ml][k0 + 1];
      b[0] = wr[k0];       b[1] = wr[k0 + 1];
      acc = wmma4(a, b, acc);
    }
    float bb = b1[ncol];
#pragma unroll
    for (int v = 0; v < 8; ++v) {
      int row = v + (lane >> 4) * 8;
      float x = acc[v] + bb;
      hid[row][ncol] = x > 0.f ? x : 0.f;
    }
  }
  __syncthreads();
  if (wave < 2) {
    int ncol = wave * 16 + ml;           // 0..31 (18 valid)
    int ncl = ncol < NCLS ? ncol : NCLS - 1;
    v8f acc = {};
    const float* wr = W2 + (size_t)ncl * 256;
    for (int k4 = 0; k4 < 64; ++k4) {
      int k0 = k4 * 4 + kb;
      v2f a, b;
      a[0] = hid[ml][k0]; a[1] = hid[ml][k0 + 1];
      float w0 = wr[k0], w1 = wr[k0 + 1];
      b[0] = ncol < NCLS ? w0 : 0.f;
      b[1] = ncol < NCLS ? w1 : 0.f;
      acc = wmma4(a, b, acc);
    }
    if (ncol < NCLS) {
      float bb = b2[ncol];
#pragma unroll
      for (int v = 0; v < 8; ++v) {
        int row = v + (lane >> 4) * 8;
        E[(n0 + row) * NCLS + ncol] = acc[v] + bb;
      }
    }
  }
}

// ===========================================================================
// CRF negative log-likelihood. Single block, 576 threads = (b,k) pairs.
// ===========================================================================
__global__ void __launch_bounds__(576)
crf_loss_kernel(const float* __restrict__ E, const int* __restrict__ tags,
                const int* __restrict__ seq_lens, const float* __restrict__ st,
                const float* __restrict__ et, const float* __restrict__ trans,
                float* __restrict__ out0) {
  __shared__ float nbuf[2][BS][NCLS];
  __shared__ float scoreb[BS];
  __shared__ float resb[BS];
  const int tid = threadIdx.x;
  const int b = tid / NCLS, k = tid % NCLS;
  const int Lb = seq_lens[b];
  nbuf[0][b][k] = st[k] + E[(b * SLEN) * NCLS + k];
  if (tid < BS) {
    int tg = tags[tid * SLEN];
    scoreb[tid] = st[tg] + E[(tid * SLEN) * NCLS + tg];
  }
  __syncthreads();
  int cur = 0;
  for (int tt = 1; tt < SLEN; ++tt) {
    const bool m = tt < Lb;
    float mx = -1e30f;
    for (int j = 0; j < NCLS; ++j) {
      float v = nbuf[cur][b][j] + trans[j * NCLS + k];
      if (v > mx) mx = v;
    }
    float s = 0.f;
    for (int j = 0; j < NCLS; ++j)
      s += expf(nbuf[cur][b][j] + trans[j * NCLS + k] - mx);
    float ns = mx + logf(s) + E[(b * SLEN + tt) * NCLS + k];
    nbuf[cur ^ 1][b][k] = m ? ns : nbuf[cur][b][k];
    if (k == 0 && m) {
      int tp = tags[b * SLEN + tt - 1], tn = tags[b * SLEN + tt];
      scoreb[b] += trans[tp * NCLS + tn] + E[(b * SLEN + tt) * NCLS + tn];
    }
    __syncthreads();
    cur ^= 1;
  }
  if (tid < BS) {
    int L = seq_lens[tid];
    float sc = scoreb[tid] + et[tags[tid * SLEN + (L - 1)]];
    float mx = -1e30f;
    for (int kk = 0; kk < NCLS; ++kk) {
      float v = nbuf[cur][tid][kk] + et[kk];
      if (v > mx) mx = v;
    }
    float s = 0.f;
    for (int kk = 0; kk < NCLS; ++kk)
      s += expf(nbuf[cur][tid][kk] + et[kk] - mx);
    resb[tid] = sc - (mx + logf(s));
  }
  __syncthreads();
  if (tid == 0) {
    float tot = 0.f; int msum = 0;
    for (int bb = 0; bb < BS; ++bb) { tot += resb[bb]; msum += seq_lens[bb]; }
    out0[0] = -(tot / (float)msum);
  }
}

// ===========================================================================
// Viterbi decode. Single block, 576 threads forward; 32 threads backtrack.
// ===========================================================================
__global__ void __launch_bounds__(576)
crf_decode_kernel(const float* __restrict__ E, const int* __restrict__ seq_lens,
                  const float* __restrict__ st, const float* __restrict__ et,
                  const float* __restrict__ trans, int* __restrict__ tlist,
                  float* __restrict__ preds) {
  __shared__ float sbuf[2][BS][NCLS];
  const int tid = threadIdx.x;
  const int b = tid / NCLS, k = tid % NCLS;
  const int Lb = seq_lens[b];
  sbuf[0][b][k] = st[k] + E[(b * SLEN) * NCLS + k];
  __syncthreads();
  int cur = 0;
  for (int tt = 1; tt < SLEN; ++tt) {
    const bool m = tt < Lb;
    float best = -1e30f; int bi = 0;
    for (int j = 0; j < NCLS; ++j) {
      float v = sbuf[cur][b][j] + trans[j * NCLS + k];
      if (v > best) { best = v; bi = j; }
    }
    float ns = best + E[(b * SLEN + tt) * NCLS + k];
    tlist[((tt - 1) * BS + b) * NCLS + k] = bi;
    sbuf[cur ^ 1][b][k] = m ? ns : sbuf[cur][b][k];
    __syncthreads();
    cur ^= 1;
  }
  if (tid < BS) {
    float best = -1e30f; int bi = 0;
    for (int kk = 0; kk < NCLS; ++kk) {
      float v = sbuf[cur][tid][kk] + et[kk];
      if (v > best) { best = v; bi = kk; }
    }
    const int L = seq_lens[tid] - 1;
    int c = bi;
    preds[tid * SLEN + (SLEN - 1)] = (float)bi;
    for (int j = SLEN - 2; j >= 0; --j) {
      int prev = tlist[(j * BS + tid) * NCLS + c];
      if (j < L) c = prev;
      preds[tid * SLEN + j] = (float)c;
    }
  }
}

// ===========================================================================
static CellP cell_from(void* const* d_in, int base) {
  CellP p;
  p.Wx   = (const float*)d_in[base + 0]; p.bx   = (const float*)d_in[base + 1];
  p.Wh   = (const float*)d_in[base + 2]; p.bh   = (const float*)d_in[base + 3];
  p.lnxg = (const float*)d_in[base + 4]; p.lnxb = (const float*)d_in[base + 5];
  p.lnhg = (const float*)d_in[base + 6]; p.lnhb = (const float*)d_in[base + 7];
  p.lncg = (const float*)d_in[base + 8]; p.lncb = (const float*)d_in[base + 9];
  return p;
}

extern "C" void kernel_launch(void* const* d_in, const int* in_sizes, int n_in,
                              void* d_out, int out_size, void* d_ws,
                              size_t ws_size, hipStream_t stream) {
  // Input flattening order (setup_inputs, depth-first):
  // 0 w_emb, 1 c_emb, 2-11 char.f, 12-21 char.b, 22-31 word.f0, 32-41 word.f1,
  // 42-51 word.b0, 52-61 word.b1, 62 W1, 63 b1, 64 W2, 65 b2,
  // 66 st, 67 et, 68 trans, 69 s, 70 c, 71 tags, 72 seq_lens
  const float* w_emb = (const float*)d_in[0];
  const float* c_emb = (const float*)d_in[1];
  CellP chf = cell_from(d_in, 2);
  CellP chb = cell_from(d_in, 12);
  CellP wf0 = cell_from(d_in, 22);
  CellP wf1 = cell_from(d_in, 32);
  CellP wb0 = cell_from(d_in, 42);
  CellP wb1 = cell_from(d_in, 52);
  const float* W1 = (const float*)d_in[62];
  const float* b1 = (const float*)d_in[63];
  const float* W2 = (const float*)d_in[64];
  const float* b2 = (const float*)d_in[65];
  const float* st = (const float*)d_in[66];
  const float* et = (const float*)d_in[67];
  const float* tr = (const float*)d_in[68];
  const int* sidx = (const int*)d_in[69];
  const int* cidx = (const int*)d_in[70];
  const int* tags = (const int*)d_in[71];
  const int* slen = (const int*)d_in[72];

  // Workspace carve-up (floats): ~15.3 MB total
  float* ws   = (float*)d_ws;
  float* hcat = ws;                        // 2*4096*64   = 524288
  float* X    = hcat + 2 * NSEQ * 64;      // 128*32*384  = 1572864
  float* OUTF = X + SLEN * BS * 384;       // 128*32*192  = 786432
  float* OUTB = OUTF + SLEN * BS * 192;    // 786432
  float* Emat = OUTB + SLEN * BS * 192;    // 4096*18     = 73728
  int*   TL   = (int*)(Emat + NSEQ * NCLS);// 127*32*18 ints

  dim3 cgrid(NSEQ / 16, 2);
  char_lstm_kernel<<<cgrid, 256, 0, stream>>>(c_emb, cidx, chf, chb, hcat);

  int bx_total = SLEN * BS * 384;
  build_x_kernel<<<(bx_total + 255) / 256, 256, 0, stream>>>(w_emb, sidx, hcat, X);

  word_lstm_kernel<<<2, 768, 0, stream>>>(X, wf0, wf1, wb0, wb1, OUTF, OUTB);

  mlp_kernel<<<NSEQ / 16, 256, 0, stream>>>(OUTF, OUTB, W1, b1, W2, b2, Emat);

  crf_loss_kernel<<<1, 576, 0, stream>>>(Emat, tags, slen, st, et, tr,
                                         (float*)d_out);
  crf_decode_kernel<<<1, 576, 0, stream>>>(Emat, slen, st, et, tr, TL,
                                           (float*)d_out + 1);
}